// EdgeConv_36739150250675
// MI455X (gfx1250) — compile-verified
//
#include <hip/hip_runtime.h>
#include <hip/hip_bf16.h>
#include <stdint.h>

// ---------------------------------------------------------------------------
// EdgeConv (DGCNN) fused pipeline for MI455X (gfx1250, wave32, WMMA).
// GEMMs use V_WMMA_F32_16X16X4_F32; feature staging uses the CDNA5 async
// global->LDS path (GLOBAL_LOAD_ASYNC_TO_LDS_B128 + s_wait_asynccnt).
// conv1 is split:  y1 = Wn @ X_nbr + Wc @ X_ctr  (Wc = W1[:,64:]-W1[:,:64]),
// and the center term is computed once per point, not once per edge.
// ---------------------------------------------------------------------------

#define BB 16
#define CC 64
#define NN 2048
#define KNB 20
#define O1 64
#define O2 128
#define MTOT (BB * NN * KNB)  // 655360 edge columns
#define G1 1024               // workgroups for conv1-stats pass
#define G2 1024               // workgroups for fused pass
#define NGROUP (BB * NN / 4)  // 8192 groups of (4 points x 20 neighbors)
#define FPITCH 68             // point-major LDS pitch (17 x 16B), conflict-free
#define EPS 1e-5f
#define SLOPE 0.2f

typedef float v2f __attribute__((ext_vector_type(2)));
typedef float v8f __attribute__((ext_vector_type(8)));

__device__ __forceinline__ v8f wmma4(v2f a, v2f b, v8f c) {
  // D(16x16,f32) += A(16x4,f32) * B(4x16,f32)
  return __builtin_amdgcn_wmma_f32_16x16x4_f32(false, a, false, b, (short)0, c,
                                               false, false);
}

// CDNA5 async copy: 16 bytes global -> LDS, tracked by ASYNCcnt.
__device__ __forceinline__ void async_copy16(unsigned lds_off,
                                             const float* gptr) {
  asm volatile("global_load_async_to_lds_b128 %0, %1, off" ::"v"(lds_off),
               "v"((unsigned long long)(uintptr_t)gptr)
               : "memory");
}
__device__ __forceinline__ void wait_async0() {
#if __has_builtin(__builtin_amdgcn_s_wait_asynccnt)
  __builtin_amdgcn_s_wait_asynccnt(0);
#else
  asm volatile("s_wait_asynccnt 0x0" ::: "memory");
#endif
}

// order-preserving float <-> uint key (for LDS atomicMax-based max over k)
__device__ __forceinline__ unsigned f2key(float f) {
  unsigned u = __float_as_uint(f);
  return (u & 0x80000000u) ? ~u : (u | 0x80000000u);
}
__device__ __forceinline__ float key2f(unsigned k) {
  unsigned u = (k & 0x80000000u) ? (k & 0x7fffffffu) : ~k;
  return __uint_as_float(u);
}

// ---------------------------------------------------------------------------
// K0a: transpose x (B,C,N) -> xT (B,N,C) (channel quads contiguous for B128
//      gathers) and per-point squared norms.
// ---------------------------------------------------------------------------
__global__ __launch_bounds__(256) void k_prep(const float* __restrict__ x,
                                              float* __restrict__ xT,
                                              float* __restrict__ xx) {
  int t = blockIdx.x * 256 + threadIdx.x;  // t = b*NN + n
  int b = t >> 11;
  int n = t & (NN - 1);
  const float* xb = x + (size_t)b * CC * NN;
  float* row = xT + (size_t)t * CC;
  float s = 0.f;
#pragma unroll 8
  for (int c = 0; c < CC; ++c) {
    float v = xb[c * NN + n];  // lanes vary n -> coalesced
    s += v * v;
    row[c] = v;
  }
  xx[t] = s;
}

// K0b: Wc = W1[:,64:] - W1[:,:64]
__global__ __launch_bounds__(256) void k_prepw(const float* __restrict__ W1,
                                               float* __restrict__ Wc) {
  int i = blockIdx.x * 256 + threadIdx.x;  // 4096
  int r = i >> 6, c = i & 63;
  Wc[i] = W1[r * 128 + 64 + c] - W1[r * 128 + c];
}

// ---------------------------------------------------------------------------
// K1: KNN. One wave per (batch, 16-row tile). Gram tiles via f32 WMMA,
//     score = 2*dot(i,j) - |xj|^2 (row term dropped: rank-invariant).
//     Per-row top-20 insertion lists in LDS (only the SET matters downstream).
// ---------------------------------------------------------------------------
__global__ __launch_bounds__(32) void k_knn(const float* __restrict__ xT,
                                            const float* __restrict__ xx,
                                            int* __restrict__ idxo) {
  int bId = blockIdx.x;  // 0..2047
  int b = bId >> 7;
  int ibase = (bId & 127) * 16;
  int lane = threadIdx.x;
  int half = lane >> 4;
  int l16 = lane & 15;

  __shared__ float stile[16 * 17];
  __shared__ float pdl[16 * 20];
  __shared__ int idl[16 * 20];

  if (lane < 16) {
#pragma unroll
    for (int t = 0; t < KNB; ++t) {
      pdl[lane * KNB + t] = -3.0e38f;
      idl[lane * KNB + t] = 0;
    }
  }

  const float* xb = xT + (size_t)b * NN * CC;
  v2f a[16];
  {
    const float* arow = xb + (size_t)(ibase + l16) * CC;
#pragma unroll
    for (int kk = 0; kk < 16; ++kk)
      a[kk] = *(const v2f*)(arow + 4 * kk + 2 * half);
  }
  __syncthreads();

  for (int jt = 0; jt < NN / 16; ++jt) {
    int jbase = jt * 16;
    const float* brow = xb + (size_t)(jbase + l16) * CC;
    v8f acc = {};
#pragma unroll
    for (int kk = 0; kk < 16; ++kk) {
      v2f bv = *(const v2f*)(brow + 4 * kk + 2 * half);
      acc = wmma4(a[kk], bv, acc);
    }
    float xxj = xx[b * NN + jbase + l16];
#pragma unroll
    for (int v = 0; v < 8; ++v)
      stile[(v + 8 * half) * 17 + l16] = 2.0f * acc[v] - xxj;
    __syncthreads();

    if (lane < 16) {  // lane r owns row r's top-20 list (sorted ascending)
      float* pdr = &pdl[lane * KNB];
      int* idr = &idl[lane * KNB];
      const float* srow = &stile[lane * 17];
      for (int jj = 0; jj < 16; ++jj) {
        float v = srow[jj];
        if (v > pdr[0]) {
          int pos = 0;
          while (pos < KNB - 1 && pdr[pos + 1] < v) {
            pdr[pos] = pdr[pos + 1];
            idr[pos] = idr[pos + 1];
            ++pos;
          }
          pdr[pos] = v;
          idr[pos] = jbase + jj;
        }
      }
    }
    __syncthreads();
  }

  if (lane < 16) {
    int* op = idxo + ((size_t)b * NN + (ibase + lane)) * KNB;
#pragma unroll
    for (int t = 0; t < KNB; ++t) op[t] = idl[lane * KNB + t];
  }
}

// ---------------------------------------------------------------------------
// K2: conv1 BN1 batch statistics. Group = 4 points x 20 neighbors.
//     Neighbor columns staged async to LDS (point-major, pitch 68);
//     center term yc = Wc @ X_ctr computed once per group per wave (oc=w),
//     broadcast into edge columns by wave shuffle. Register stats,
//     deterministic per-WG partials.
// ---------------------------------------------------------------------------
__global__ __launch_bounds__(128) void k_conv1_stats(
    const float* __restrict__ xT, const int* __restrict__ idxi,
    const float* __restrict__ W1, const float* __restrict__ Wc,
    float* __restrict__ part1) {
  __shared__ __align__(16) float feat[80 * FPITCH];  // 21.25 KB
  int tid = threadIdx.x;
  int w = tid >> 5, lane = tid & 31, half = lane >> 4, l16 = lane & 15;
  int oc = w;  // fixed 16-channel stripe per wave

  v2f a[16], ac[16];
  {
    const float* wnrow = W1 + (oc * 16 + l16) * 128;
    const float* wcrow = Wc + (oc * 16 + l16) * 64;
#pragma unroll
    for (int kk = 0; kk < 16; ++kk) {
      a[kk] = *(const v2f*)(wnrow + 4 * kk + 2 * half);
      ac[kk] = *(const v2f*)(wcrow + 4 * kk + 2 * half);
    }
  }
  unsigned featBase = (unsigned)(uintptr_t)feat;
  float sumA[8] = {0, 0, 0, 0, 0, 0, 0, 0};
  float sqA[8] = {0, 0, 0, 0, 0, 0, 0, 0};

  for (int grp = blockIdx.x; grp < NGROUP; grp += gridDim.x) {
    int b = grp >> 9;
    int n0 = (grp & 511) * 4;
    // async-stage 80 neighbor columns x 64 channels (16B per task)
#pragma unroll
    for (int it = 0; it < 10; ++it) {
      int task = it * 128 + tid;
      int p = task >> 4, s = task & 15;
      int n = n0 + p / KNB, j = p % KNB;
      int nbr = idxi[((size_t)b * NN + n) * KNB + j];
      async_copy16(featBase + (unsigned)(p * (FPITCH * 4) + s * 16),
                   xT + ((size_t)b * NN + nbr) * CC + 4 * s);
    }
    // yc = Wc @ X_ctr (cols 0..3 valid; others clamped duplicates)
    v8f yc = {};
    {
      int nc = n0 + (l16 < 4 ? l16 : 3);
      const float* crow = xT + ((size_t)b * NN + nc) * CC;
#pragma unroll
      for (int kk = 0; kk < 16; ++kk) {
        v2f bv = *(const v2f*)(crow + 4 * kk + 2 * half);
        yc = wmma4(ac[kk], bv, yc);
      }
    }
    wait_async0();
    __syncthreads();
#pragma unroll
    for (int pt = 0; pt < 5; ++pt) {
      int pb = pt * 16;
      const float* frow = feat + (pb + l16) * FPITCH;
      v8f acc = {};
#pragma unroll
      for (int kk = 0; kk < 16; ++kk) {
        v2f bv = *(const v2f*)(frow + 4 * kk + 2 * half);
        acc = wmma4(a[kk], bv, acc);
      }
      int src = (lane & 16) | ((pb + l16) / KNB);  // same-half shuffle source
#pragma unroll
      for (int v = 0; v < 8; ++v) {
        float y = acc[v] + __shfl(yc[v], src, 32);
        sumA[v] += y;
        sqA[v] += y * y;
      }
    }
    __syncthreads();
  }
#pragma unroll
  for (int v = 0; v < 8; ++v) {
    float s = sumA[v], q = sqA[v];
    for (int mk = 8; mk >= 1; mk >>= 1) {
      s += __shfl_xor(s, mk, 32);
      q += __shfl_xor(q, mk, 32);
    }
    if (l16 == 0) {
      int ch = oc * 16 + v + 8 * half;
      part1[((size_t)blockIdx.x * O1 + ch) * 2 + 0] = s;
      part1[((size_t)blockIdx.x * O1 + ch) * 2 + 1] = q;
    }
  }
}

__global__ void k_bn1(const float* __restrict__ part1,
                      const float* __restrict__ g, const float* __restrict__ b,
                      float* __restrict__ bnp1) {
  int ch = threadIdx.x;  // 64
  float S = 0.f, Q = 0.f;
  for (int wg = 0; wg < G1; ++wg) {
    S += part1[((size_t)wg * O1 + ch) * 2 + 0];
    Q += part1[((size_t)wg * O1 + ch) * 2 + 1];
  }
  const float Mf = (float)MTOT;
  float mu = S / Mf;
  float var = Q / Mf - mu * mu;
  float sc = g[ch] * rsqrtf(var + EPS);
  bnp1[ch] = sc;
  bnp1[O1 + ch] = b[ch] - mu * sc;
}

// ---------------------------------------------------------------------------
// K4: fused conv1 -> BN1+LReLU -> conv2, max over k, BN2 stats.
//     8 waves; waves 0-3 compute the per-point center term into LDS while
//     the neighbor tile streams in asynchronously; conv2 channel stripe is
//     wave-fixed (register stats). Max over k via LDS atomicMax on keys.
// ---------------------------------------------------------------------------
__global__ __launch_bounds__(256) void k_fused2(
    const float* __restrict__ xT, const int* __restrict__ idxi,
    const float* __restrict__ W1, const float* __restrict__ Wc,
    const float* __restrict__ bnp1, const float* __restrict__ W2,
    float* __restrict__ part2, float* __restrict__ out) {
  __shared__ __align__(16) float feat[80 * FPITCH];  // 21.25 KB
  __shared__ v2f h1g[32 * 80];                       // 20 KB
  __shared__ float ycb[O1 * 4];                      // 1 KB center terms
  __shared__ unsigned maxk[O2 * 4];                  // 2 KB maxima keys

  int tid = threadIdx.x;
  int w = tid >> 5, lane = tid & 31, half = lane >> 4, l16 = lane & 15;
  int oc2 = w;

  v2f a2[16];
  {
    const float* wrow = W2 + (oc2 * 16 + l16) * 64;
#pragma unroll
    for (int kk = 0; kk < 16; ++kk)
      a2[kk] = *(const v2f*)(wrow + 4 * kk + 2 * half);
  }
  v2f ac[16];
  if (w < 4) {  // wave-uniform: waves 0-3 own conv1 center stripes
    const float* wcrow = Wc + (w * 16 + l16) * 64;
#pragma unroll
    for (int kk = 0; kk < 16; ++kk)
      ac[kk] = *(const v2f*)(wcrow + 4 * kk + 2 * half);
  }
  unsigned featBase = (unsigned)(uintptr_t)feat;
  float sumA[8] = {0, 0, 0, 0, 0, 0, 0, 0};
  float sqA[8] = {0, 0, 0, 0, 0, 0, 0, 0};

  for (int grp = blockIdx.x; grp < NGROUP; grp += gridDim.x) {
    int b = grp >> 9;
    int n0 = (grp & 511) * 4;
    maxk[tid] = 0u;
    maxk[tid + 256] = 0u;
    // async-stage 80 neighbor columns (1280 x 16B)
#pragma unroll
    for (int it = 0; it < 5; ++it) {
      int task = it * 256 + tid;
      int p = task >> 4, s = task & 15;
      int n = n0 + p / KNB, j = p % KNB;
      int nbr = idxi[((size_t)b * NN + n) * KNB + j];
      async_copy16(featBase + (unsigned)(p * (FPITCH * 4) + s * 16),
                   xT + ((size_t)b * NN + nbr) * CC + 4 * s);
    }
    // center term while the async copies are in flight
    if (w < 4) {
      int nc = n0 + (l16 < 4 ? l16 : 3);
      const float* crow = xT + ((size_t)b * NN + nc) * CC;
      v8f yc = {};
#pragma unroll
      for (int kk = 0; kk < 16; ++kk) {
        v2f bv = *(const v2f*)(crow + 4 * kk + 2 * half);
        yc = wmma4(ac[kk], bv, yc);
      }
#pragma unroll
      for (int v = 0; v < 8; ++v)
        if (l16 < 4) ycb[(w * 16 + v + 8 * half) * 4 + l16] = yc[v];
    }
    wait_async0();
    __syncthreads();

    // conv1 neighbor part + center add + BN1 + LReLU -> h1g (20 tile-tasks)
    for (int t = w; t < 20; t += 8) {
      int oc1 = t & 3, pb = (t >> 2) * 16;
      const float* wnrow = W1 + (oc1 * 16 + l16) * 128;
      const float* frow = feat + (pb + l16) * FPITCH;
      v8f acc = {};
#pragma unroll
      for (int kk = 0; kk < 16; ++kk) {
        v2f av = *(const v2f*)(wnrow + 4 * kk + 2 * half);
        v2f bv = *(const v2f*)(frow + 4 * kk + 2 * half);
        acc = wmma4(av, bv, acc);
      }
      int nl = (pb + l16) / KNB;
      float h[8];
#pragma unroll
      for (int v = 0; v < 8; ++v) {
        int ch = oc1 * 16 + v + 8 * half;
        float y = acc[v] + ycb[ch * 4 + nl];
        y = y * bnp1[ch] + bnp1[O1 + ch];
        h[v] = (y >= 0.f) ? y : SLOPE * y;
      }
#pragma unroll
      for (int v2 = 0; v2 < 4; ++v2) {
        int cp = oc1 * 8 + 4 * half + v2;
        v2f hv;
        hv.x = h[2 * v2];
        hv.y = h[2 * v2 + 1];
        h1g[cp * 80 + pb + l16] = hv;
      }
    }
    __syncthreads();

    // conv2 (wave-fixed channel stripe) + stats + max over k
    for (int pt = 0; pt < 5; ++pt) {
      int pb = pt * 16;
      v8f acc = {};
#pragma unroll
      for (int kk = 0; kk < 16; ++kk) {
        v2f bv = h1g[(2 * kk + half) * 80 + pb + l16];
        acc = wmma4(a2[kk], bv, acc);
      }
      int nl = (pb + l16) / KNB;
#pragma unroll
      for (int v = 0; v < 8; ++v) {
        float y = acc[v];
        sumA[v] += y;
        sqA[v] += y * y;
        int ch = oc2 * 16 + v + 8 * half;
        atomicMax(&maxk[ch * 4 + nl], f2key(y));
      }
    }
    __syncthreads();

    // raw (pre-BN2) maxima -> d_out; BN2+LReLU applied later (monotone map)
    for (int i = tid; i < O2 * 4; i += 256) {
      int ch = i >> 2, nl = i & 3;
      out[((size_t)b * O2 + ch) * NN + n0 + nl] = key2f(maxk[i]);
    }
    __syncthreads();
  }

#pragma unroll
  for (int v = 0; v < 8; ++v) {
    float s = sumA[v], q = sqA[v];
    for (int mk = 8; mk >= 1; mk >>= 1) {
      s += __shfl_xor(s, mk, 32);
      q += __shfl_xor(q, mk, 32);
    }
    if (l16 == 0) {
      int ch = oc2 * 16 + v + 8 * half;
      part2[((size_t)blockIdx.x * O2 + ch) * 2 + 0] = s;
      part2[((size_t)blockIdx.x * O2 + ch) * 2 + 1] = q;
    }
  }
}

__global__ void k_bn2(const float* __restrict__ part2,
                      const float* __restrict__ g, const float* __restrict__ b,
                      float* __restrict__ bnp2) {
  int ch = threadIdx.x;  // 128
  float S = 0.f, Q = 0.f;
  for (int wg = 0; wg < G2; ++wg) {
    S += part2[((size_t)wg * O2 + ch) * 2 + 0];
    Q += part2[((size_t)wg * O2 + ch) * 2 + 1];
  }
  const float Mf = (float)MTOT;
  float mu = S / Mf;
  float var = Q / Mf - mu * mu;
  float sc = g[ch] * rsqrtf(var + EPS);
  bnp2[ch] = sc;
  bnp2[O2 + ch] = b[ch] - mu * sc;
}

// BN2 + LeakyReLU after the max (valid: positive scale => monotone map).
__global__ __launch_bounds__(256) void k_final(float* __restrict__ out,
                                               const float* __restrict__ bnp2) {
  const int total = BB * O2 * NN;
  for (int i = blockIdx.x * 256 + threadIdx.x; i < total;
       i += gridDim.x * 256) {
    int ch = (i >> 11) & (O2 - 1);
    float y = out[i] * bnp2[ch] + bnp2[O2 + ch];
    out[i] = (y >= 0.f) ? y : SLOPE * y;
  }
}

// ---------------------------------------------------------------------------
extern "C" void kernel_launch(void* const* d_in, const int* in_sizes, int n_in,
                              void* d_out, int out_size, void* d_ws,
                              size_t ws_size, hipStream_t stream) {
  const float* x = (const float*)d_in[0];
  const float* W1 = (const float*)d_in[1];
  const float* g1 = (const float*)d_in[2];
  const float* b1 = (const float*)d_in[3];
  const float* W2 = (const float*)d_in[4];
  const float* g2 = (const float*)d_in[5];
  const float* b2 = (const float*)d_in[6];
  float* out = (float*)d_out;

  char* ws = (char*)d_ws;
  float* xT = (float*)(ws + 0);            //  8,388,608 B
  float* xx = (float*)(ws + 8388608);      //    131,072 B
  int* idxb = (int*)(ws + 8519680);        //  2,621,440 B
  float* part1 = (float*)(ws + 11141120);  //    524,288 B
  float* bnp1 = (float*)(ws + 11665408);   //        512 B
  float* part2 = (float*)(ws + 11665920);  //  1,048,576 B
  float* bnp2 = (float*)(ws + 12714496);   //      1,024 B
  float* Wc = (float*)(ws + 12715520);     //     16,384 B

  k_prep<<<BB * NN / 256, 256, 0, stream>>>(x, xT, xx);
  k_prepw<<<16, 256, 0, stream>>>(W1, Wc);
  k_knn<<<BB * (NN / 16), 32, 0, stream>>>(xT, xx, idxb);
  k_conv1_stats<<<G1, 128, 0, stream>>>(xT, idxb, W1, Wc, part1);
  k_bn1<<<1, O1, 0, stream>>>(part1, g1, b1, bnp1);
  k_fused2<<<G2, 256, 0, stream>>>(xT, idxb, W1, Wc, bnp1, W2, part2, out);
  k_bn2<<<1, O2, 0, stream>>>(part2, g2, b2, bnp2);
  k_final<<<2048, 256, 0, stream>>>(out, bnp2);
}